// GCN_47227460387480
// MI455X (gfx1250) — compile-verified
//
#include <hip/hip_runtime.h>

typedef __attribute__((ext_vector_type(2))) float v2f;
typedef __attribute__((ext_vector_type(8))) float v8f;

#define N_NODES   100000
#define F_INPUT   128
#define HID       32            // H
#define E_EDGES   1600000
#define G_GRAPHS  128
#define C_CLASSES 10

// ---------------------------------------------------------------- utilities
__global__ void fill_kernel(float* __restrict__ p, float v, int n) {
    int i = blockIdx.x * blockDim.x + threadIdx.x;
    if (i < n) p[i] = v;
}

// deg[dst] += 1 per edge (deg buffers pre-filled with 1.0 for the self loop)
__global__ void deg_kernel(const int* __restrict__ dst, float* __restrict__ deg) {
    int i = blockIdx.x * blockDim.x + threadIdx.x;
    if (i < E_EDGES) atomicAdd(&deg[dst[i]], 1.0f);
}

__global__ void rsqrt_kernel(float* __restrict__ d, int n) {
    int i = blockIdx.x * blockDim.x + threadIdx.x;
    if (i < n) d[i] = rsqrtf(d[i]);
}

// ------------------------------------------------- WMMA GEMM 1: h1 = x @ W1
// [100000,128] @ [128,32] -> [100000,32].  One wave per 16x16 output tile.
__global__ void gemm_xW1(const float* __restrict__ x, const float* __restrict__ W1,
                         float* __restrict__ h1) {
    const int wave = threadIdx.x >> 5;
    const int lane = threadIdx.x & 31;
    const int mt   = blockIdx.x * 2 + wave;   // 0..6249
    const int nt   = blockIdx.y;              // 0..1
    const int half = lane >> 4;               // 0: K=k0,k0+1   1: K=k0+2,k0+3
    const int lr   = lane & 15;
    const int row  = mt * 16 + lr;
    const int col  = nt * 16 + lr;

    const float* __restrict__ arow = x + (long long)row * F_INPUT + 2 * half;
    v8f c = {};
    #pragma unroll 4
    for (int k0 = 0; k0 < F_INPUT; k0 += 4) {
        v2f a, b;
        a.x = arow[k0];
        a.y = arow[k0 + 1];
        const int kb = k0 + 2 * half;
        b.x = W1[kb * HID + col];
        b.y = W1[(kb + 1) * HID + col];
        c = __builtin_amdgcn_wmma_f32_16x16x4_f32(
                false, a, false, b, (short)0, c, false, false);
    }
    float* __restrict__ out = h1 + ((long long)mt * 16 + 8 * half) * HID + nt * 16 + lr;
    #pragma unroll
    for (int r = 0; r < 8; ++r) out[(long long)r * HID] = c[r];
}

// ------------------------------------- WMMA GEMM 2: h2 = [agg1|agg2] @ W2
__global__ void gemm_hW2(const float* __restrict__ agg1, const float* __restrict__ agg2,
                         const float* __restrict__ W2, float* __restrict__ h2) {
    const int wave = threadIdx.x >> 5;
    const int lane = threadIdx.x & 31;
    const int mt   = blockIdx.x * 2 + wave;   // 0..6249
    const int nt   = blockIdx.y;              // 0..3
    const int half = lane >> 4;
    const int lr   = lane & 15;
    const int row  = mt * 16 + lr;
    const int col  = nt * 16 + lr;

    v8f c = {};
    #pragma unroll
    for (int k0 = 0; k0 < 64; k0 += 4) {
        const int ka = k0 + 2 * half;         // ka, ka+1 stay inside one 32-half
        const float* __restrict__ src = (ka < 32)
            ? (agg1 + (long long)row * HID + ka)
            : (agg2 + (long long)row * HID + (ka - 32));
        v2f a, b;
        a.x = src[0];
        a.y = src[1];
        b.x = W2[ka * 64 + col];
        b.y = W2[(ka + 1) * 64 + col];
        c = __builtin_amdgcn_wmma_f32_16x16x4_f32(
                false, a, false, b, (short)0, c, false, false);
    }
    float* __restrict__ out = h2 + ((long long)mt * 16 + 8 * half) * 64 + nt * 16 + lr;
    #pragma unroll
    for (int r = 0; r < 8; ++r) out[(long long)r * 64] = c[r];
}

// ------------------------------------------------------------- edge scatter
// One wave per 4 edges; e is wave-uniform (readfirstlane) so ei[]/dis[] reads
// become scalar (SMEM) loads.  Lane = feature.  Grid sized exactly, no tail.
__global__ void scatter32_kernel(const int* __restrict__ ei, const float* __restrict__ dis,
                                 const float* __restrict__ h, float* __restrict__ agg) {
    const int lane = threadIdx.x & 31;
    int wid = blockIdx.x * (blockDim.x >> 5) + (threadIdx.x >> 5);
    wid = __builtin_amdgcn_readfirstlane(wid);
    const int ebase = wid * 4;
    #pragma unroll
    for (int j = 0; j < 4; ++j) {
        const int e   = ebase + j;
        const int src = ei[e];
        const int dst = ei[E_EDGES + e];
        const float nrm = dis[src] * dis[dst];
        atomicAdd(&agg[(long long)dst * HID + lane],
                  h[(long long)src * HID + lane] * nrm);
    }
}

// 64-feature variant: lane covers 2 consecutive floats (float2 gather).
__global__ void scatter64_kernel(const int* __restrict__ ei, const float* __restrict__ dis,
                                 const float* __restrict__ h, float* __restrict__ agg) {
    const int lane = threadIdx.x & 31;
    int wid = blockIdx.x * (blockDim.x >> 5) + (threadIdx.x >> 5);
    wid = __builtin_amdgcn_readfirstlane(wid);
    const int ebase = wid * 4;
    #pragma unroll
    for (int j = 0; j < 4; ++j) {
        const int e   = ebase + j;
        const int src = ei[e];
        const int dst = ei[E_EDGES + e];
        const float nrm = dis[src] * dis[dst];
        const float2 hv = *(const float2*)(h + (long long)src * 64 + 2 * lane);
        float* ap = agg + (long long)dst * 64 + 2 * lane;
        atomicAdd(ap,     hv.x * nrm);
        atomicAdd(ap + 1, hv.y * nrm);
    }
}

// ---------------- layer-1 finalize: agg = relu(agg + h*dis^2 + bias) in place
__global__ void finalize32_kernel(float* __restrict__ agg, const float* __restrict__ h,
                                  const float* __restrict__ dis, const float* __restrict__ bias) {
    long long t = (long long)blockIdx.x * blockDim.x + threadIdx.x;
    if (t >= (long long)N_NODES * HID) return;
    long long i = t >> 5;
    int f = (int)(t & 31);
    float d = dis[i];
    float v = agg[t] + h[t] * d * d + bias[f];
    agg[t] = v > 0.0f ? v : 0.0f;
}

// --------- pooling fused with layer-2 finalize (self loop + bias + ReLU)
__global__ void pool_kernel(const float* __restrict__ agg3, const float* __restrict__ agg4,
                            const float* __restrict__ h2pre,
                            const float* __restrict__ dis1, const float* __restrict__ dis2,
                            const float* __restrict__ b2, const int* __restrict__ batch,
                            float* __restrict__ pooled, float* __restrict__ cnt) {
    long long t = (long long)blockIdx.x * blockDim.x + threadIdx.x; // N*128
    if (t >= (long long)N_NODES * 128) return;
    long long i = t >> 7;
    int f = (int)(t & 127);
    int g = batch[i];
    float v;
    if (f < 64) {
        float d = dis1[i];
        v = agg3[i * 64 + f] + h2pre[i * 64 + f] * d * d + b2[f];
    } else {
        int f2 = f - 64;
        float d = dis2[i];
        v = agg4[i * 64 + f2] + h2pre[i * 64 + f2] * d * d + b2[f2];
    }
    v = fmaxf(v, 0.0f);
    atomicAdd(&pooled[g * 128 + f], v);
    if (f == 0) atomicAdd(&cnt[g], 1.0f);
}

// ------------------------------------------------- final FC: out = mean @ Wfc
__global__ void fc_kernel(const float* __restrict__ pooled, const float* __restrict__ cnt,
                          const float* __restrict__ Wfc, const float* __restrict__ bfc,
                          float* __restrict__ out) {
    int t = blockIdx.x * blockDim.x + threadIdx.x;
    if (t >= G_GRAPHS * C_CLASSES) return;
    int g = t / C_CLASSES;
    int c = t % C_CLASSES;
    float inv = 1.0f / fmaxf(cnt[g], 1.0f);
    float acc = bfc[c];
    #pragma unroll 4
    for (int k = 0; k < 128; ++k)
        acc += pooled[g * 128 + k] * inv * Wfc[k * C_CLASSES + c];
    out[t] = acc;
}

// ================================================================== launch
extern "C" void kernel_launch(void* const* d_in, const int* in_sizes, int n_in,
                              void* d_out, int out_size, void* d_ws, size_t ws_size,
                              hipStream_t stream) {
    const float* x   = (const float*)d_in[0];
    const int*   ei1 = (const int*)  d_in[1];
    const int*   ei2 = (const int*)  d_in[2];
    const int*   bat = (const int*)  d_in[3];
    const float* W1  = (const float*)d_in[4];
    const float* b1  = (const float*)d_in[5];
    const float* W2  = (const float*)d_in[6];
    const float* b2  = (const float*)d_in[7];
    const float* Wfc = (const float*)d_in[8];
    const float* bfc = (const float*)d_in[9];
    float* out = (float*)d_out;

    // Workspace layout (floats), aggB reuses dead h1/agg1/agg2 space:
    //  [0,N) dis1   [N,2N) dis2
    //  [2N,34N) h1  [34N,66N) agg1  [66N,98N) agg2
    //  [2N,66N) agg3 (reuse)        [66N,130N) agg4 (reuse)
    //  [130N,194N) h2pre            [194N,..) pooled[G*128], cnt[G]
    const long long NN = N_NODES;
    float* W      = (float*)d_ws;
    float* dis1   = W;
    float* dis2   = W + NN;
    float* h1     = W + 2 * NN;
    float* agg1   = W + 34 * NN;
    float* agg2   = W + 66 * NN;
    float* agg3   = W + 2 * NN;
    float* agg4   = W + 66 * NN;
    float* h2pre  = W + 130 * NN;
    float* pooled = W + 194 * NN;
    float* cnt    = pooled + G_GRAPHS * 128;

    const int TB = 256;
    auto blks = [](long long n, int tb) { return (unsigned)((n + tb - 1) / tb); };
    const unsigned SCAT_BLOCKS = E_EDGES / (8 * 4);   // 8 waves/block * 4 edges/wave

    // 1) degrees -> dis = rsqrt(deg)  (init 1.0 covers the self loop)
    fill_kernel<<<blks(2 * NN, TB), TB, 0, stream>>>(dis1, 1.0f, (int)(2 * NN));
    deg_kernel<<<blks(E_EDGES, TB), TB, 0, stream>>>(ei1 + E_EDGES, dis1);
    deg_kernel<<<blks(E_EDGES, TB), TB, 0, stream>>>(ei2 + E_EDGES, dis2);
    rsqrt_kernel<<<blks(2 * NN, TB), TB, 0, stream>>>(dis1, (int)(2 * NN));

    // 2) h1 = x @ W1 (WMMA fp32), shared by both layer-1 convs
    gemm_xW1<<<dim3(3125, 2), 64, 0, stream>>>(x, W1, h1);

    // 3) layer-1 aggregation
    fill_kernel<<<blks(64 * NN, TB), TB, 0, stream>>>(agg1, 0.0f, (int)(64 * NN));
    scatter32_kernel<<<SCAT_BLOCKS, TB, 0, stream>>>(ei1, dis1, h1, agg1);
    scatter32_kernel<<<SCAT_BLOCKS, TB, 0, stream>>>(ei2, dis2, h1, agg2);
    finalize32_kernel<<<blks(32 * NN, TB), TB, 0, stream>>>(agg1, h1, dis1, b1);
    finalize32_kernel<<<blks(32 * NN, TB), TB, 0, stream>>>(agg2, h1, dis2, b1);

    // 4) h2pre = concat(agg1,agg2) @ W2 (WMMA fp32), shared by both layer-2 convs
    gemm_hW2<<<dim3(3125, 4), 64, 0, stream>>>(agg1, agg2, W2, h2pre);

    // 5) layer-2 aggregation (agg3/agg4 reuse the now-dead h1/agg1/agg2 region)
    fill_kernel<<<blks(128 * NN, TB), TB, 0, stream>>>(agg3, 0.0f, (int)(128 * NN));
    scatter64_kernel<<<SCAT_BLOCKS, TB, 0, stream>>>(ei1, dis1, h2pre, agg3);
    scatter64_kernel<<<SCAT_BLOCKS, TB, 0, stream>>>(ei2, dis2, h2pre, agg4);

    // 6) global mean pool (fused with layer-2 finalize) + FC head
    fill_kernel<<<blks(G_GRAPHS * 128 + G_GRAPHS, TB), TB, 0, stream>>>(
        pooled, 0.0f, G_GRAPHS * 128 + G_GRAPHS);
    pool_kernel<<<blks(128 * NN, TB), TB, 0, stream>>>(agg3, agg4, h2pre, dis1, dis2,
                                                       b2, bat, pooled, cnt);
    fc_kernel<<<blks(G_GRAPHS * C_CLASSES, TB), TB, 0, stream>>>(pooled, cnt, Wfc, bfc, out);
}